// DetProposalRelationHead_12979391168954
// MI455X (gfx1250) — compile-verified
//
#include <hip/hip_runtime.h>
#include <hip/hip_bf16.h>
#include <math.h>

#define B_    8
#define N_    100
#define D_    1024
#define P_    9900
#define M_    (B_ * P_)     // 79200
#define K3_   3072
#define H_    1024
#define C_    51
#define CPAD_ 64
#define TOPK_ 100

// d_out flat offsets (floats), return order: logits, probs, top_labels, top_probs, top_vals, top_idx
#define OUT_LOGITS 0
#define OUT_PROBS  (M_ * C_)            // 4039200
#define OUT_TLAB   (2 * M_ * C_)        // 8078400
#define OUT_TPROB  (OUT_TLAB + B_ * TOPK_)
#define OUT_TVAL   (OUT_TPROB + B_ * TOPK_)
#define OUT_TIDX   (OUT_TVAL + B_ * TOPK_)

typedef __attribute__((ext_vector_type(2))) float v2f;
typedef __attribute__((ext_vector_type(8))) float v8f;

// Low 32 bits of a generic pointer to a __shared__ object are the LDS byte
// offset on AMDGPU (as(3)->generic addrspacecast places the aperture in the
// high dword), which is exactly what the async-to-LDS VDST operand needs.
__device__ __forceinline__ unsigned int lds_off(const void* p) {
  return (unsigned int)(unsigned long long)p;
}

// 32 bytes (8 floats) per thread per chunk: two b128 async copies.
// INST_OFFSET is added to BOTH the LDS and the global address (ISA 10.7).
__device__ __forceinline__ void async_copy_32B(unsigned int lds_byte,
                                               const float* src) {
  unsigned long long g = (unsigned long long)src;
  asm volatile("global_load_async_to_lds_b128 %0, %1, off"
               :: "v"(lds_byte), "v"(g) : "memory");
  asm volatile("global_load_async_to_lds_b128 %0, %1, off offset:16"
               :: "v"(lds_byte), "v"(g) : "memory");
}

__device__ __forceinline__ void wait_async0() {
  asm volatile("s_wait_asynccnt 0x0" ::: "memory");
}

// ---------------- Kernel 0: pad W2 (1024x51) -> W2p (1024x64, zero tail) ----
__global__ __launch_bounds__(256) void k_pad_w2(const float* __restrict__ W2,
                                                float* __restrict__ W2p) {
  int idx = blockIdx.x * 256 + threadIdx.x;
  if (idx >= H_ * CPAD_) return;
  int k = idx >> 6, c = idx & 63;
  W2p[idx] = (c < C_) ? W2[k * C_ + c] : 0.0f;
}

// ---------------- Kernel 1: gather + GEMM1 (f32 WMMA) + bias + leaky_relu ---
// Block: 512 threads (16 waves). Tile: 128 rows x 512 cols, K = 3072.
// Wave w: m-tiles {(w>>2)*32 .. +31}, n-cols [(w&3)*128 .. +127].
// A chunks double-buffered in LDS via GLOBAL_LOAD_ASYNC_TO_LDS_B128.
#define KC  32
#define NCH (K3_ / KC)   // 96
#define AST 36           // row stride (floats): 144B keeps 16B alignment
__global__ __launch_bounds__(512) void k_gemm1(
    const float* __restrict__ inst, const float* __restrict__ phr,
    const float* __restrict__ W1, const float* __restrict__ b1,
    const int* __restrict__ conn, const int* __restrict__ pidx,
    float* __restrict__ h) {
  __shared__ float As[2][128][AST];

  const int tid  = threadIdx.x;
  const int lane = tid & 31;
  const int w    = tid >> 5;       // wave 0..15
  const int hl   = lane >> 4;      // half-wave 0/1
  const int l15  = lane & 15;
  const int mbase = blockIdx.x * 128;
  const int nbase = blockIdx.y * 512 + (w & 3) * 128;
  const int mwave = (w >> 2) * 32;

  // A staging assignment: 4 threads per row, 8 floats (32B) each
  const int arow = tid >> 2;        // 0..127
  const int acq  = (tid & 3) * 8;   // col start within chunk
  int r = mbase + arow;
  if (r >= M_) r = M_ - 1;
  const int b = r / P_, p = r % P_;
  const int si = conn[(b * 2 + 0) * P_ + p];
  const int oi = conn[(b * 2 + 1) * P_ + p];
  const int ph = pidx[b * P_ + p];
  const float* base0 = inst + (size_t)(b * N_ + si) * D_;
  const float* base1 = inst + (size_t)(b * N_ + oi) * D_;
  const float* base2 = phr  + (size_t)(b * P_ + ph) * D_;

  const unsigned int lds0 = lds_off(&As[0][arow][acq]);
  const unsigned int lds1 = lds_off(&As[1][arow][acq]);

  auto issue = [&](int k0, int bufi) {
    const int seg = k0 >> 10;  // KC divides 1024: chunks never straddle segments
    const float* src =
        (seg == 0 ? base0 : (seg == 1 ? base1 : base2)) + (k0 & 1023) + acq;
    async_copy_32B(bufi ? lds1 : lds0, src);
  };

  v8f acc[2][8];
#pragma unroll
  for (int mi = 0; mi < 2; ++mi)
#pragma unroll
    for (int ni = 0; ni < 8; ++ni)
#pragma unroll
      for (int j = 0; j < 8; ++j) acc[mi][ni][j] = 0.0f;

  issue(0, 0);  // prime the pipeline

  for (int c = 0; c < NCH; ++c) {
    const int bufi = c & 1;
    wait_async0();       // this thread's pending chunk is fully in LDS
    __syncthreads();     // all threads' pieces visible; prior compute done
    if (c + 1 < NCH) issue((c + 1) * KC, bufi ^ 1);  // overlap next fill
    const int k0 = c * KC;

#pragma unroll
    for (int kk = 0; kk < KC; kk += 4) {
      v2f a0, a1;
      a0.x = As[bufi][mwave + l15][kk + 2 * hl];
      a0.y = As[bufi][mwave + l15][kk + 2 * hl + 1];
      a1.x = As[bufi][mwave + 16 + l15][kk + 2 * hl];
      a1.y = As[bufi][mwave + 16 + l15][kk + 2 * hl + 1];
      const float* wrow = W1 + (size_t)(k0 + kk + 2 * hl) * H_ + nbase + l15;
#pragma unroll
      for (int ni = 0; ni < 8; ++ni) {
        v2f bf;
        bf.x = wrow[ni * 16];
        bf.y = wrow[H_ + ni * 16];
        acc[0][ni] = __builtin_amdgcn_wmma_f32_16x16x4_f32(
            false, a0, false, bf, (short)0, acc[0][ni], false, false);
        acc[1][ni] = __builtin_amdgcn_wmma_f32_16x16x4_f32(
            false, a1, false, bf, (short)0, acc[1][ni], false, false);
      }
    }
  }

  // Epilogue: bias + leaky_relu(0.01), write h
#pragma unroll
  for (int ni = 0; ni < 8; ++ni) {
    const int col = nbase + ni * 16 + l15;
    const float bias = b1[col];
#pragma unroll
    for (int mi = 0; mi < 2; ++mi) {
#pragma unroll
      for (int j = 0; j < 8; ++j) {
        const int row = mbase + mwave + mi * 16 + hl * 8 + j;
        if (row < M_) {
          float v = acc[mi][ni][j] + bias;
          h[(size_t)row * H_ + col] = (v > 0.0f) ? v : 0.01f * v;
        }
      }
    }
  }
}

// ---------------- Kernel 2: GEMM2 (f32 WMMA) + softmax + pred + overall -----
// Block: 256 threads (8 waves). 128 rows per block, one 16-row m-tile per wave.
__global__ __launch_bounds__(256) void k_gemm2(
    const float* __restrict__ h, const float* __restrict__ W2p,
    const float* __restrict__ b2, const float* __restrict__ scores,
    const int* __restrict__ conn, float* __restrict__ out,
    float* __restrict__ ov, float* __restrict__ pp, float* __restrict__ pl) {
  __shared__ float Ls[8][16][CPAD_];

  const int tid  = threadIdx.x;
  const int lane = tid & 31;
  const int w    = tid >> 5;
  const int hl   = lane >> 4;
  const int l15  = lane & 15;
  const int rbase = blockIdx.x * 128 + w * 16;
  int rload = rbase + l15;
  if (rload >= M_) rload = M_ - 1;
  const float* hrow = h + (size_t)rload * H_;

  v8f acc[4];
#pragma unroll
  for (int ni = 0; ni < 4; ++ni)
#pragma unroll
    for (int j = 0; j < 8; ++j) acc[ni][j] = 0.0f;

  for (int k = 0; k < H_; k += 4) {
    v2f a;
    a.x = hrow[k + 2 * hl];
    a.y = hrow[k + 2 * hl + 1];
    const float* wrow = W2p + (size_t)(k + 2 * hl) * CPAD_ + l15;
#pragma unroll
    for (int ni = 0; ni < 4; ++ni) {
      v2f bf;
      bf.x = wrow[ni * 16];
      bf.y = wrow[CPAD_ + ni * 16];
      acc[ni] = __builtin_amdgcn_wmma_f32_16x16x4_f32(
          false, a, false, bf, (short)0, acc[ni], false, false);
    }
  }

  // dump logits (+bias) to LDS
#pragma unroll
  for (int ni = 0; ni < 4; ++ni) {
    const int col = ni * 16 + l15;
    const float bias = (col < C_) ? b2[col] : 0.0f;
#pragma unroll
    for (int j = 0; j < 8; ++j) Ls[w][hl * 8 + j][col] = acc[ni][j] + bias;
  }
  __syncthreads();

  if (lane < 16) {
    const int rr = rbase + lane;
    if (rr < M_) {
      float* lrow = &Ls[w][lane][0];
      float m = lrow[0];
      for (int c = 1; c < C_; ++c) m = fmaxf(m, lrow[c]);
      float sum = 0.0f;
      float* olog = out + OUT_LOGITS + (size_t)rr * C_;
      for (int c = 0; c < C_; ++c) {
        float lg = lrow[c];
        olog[c] = lg;
        float e = __expf(lg - m);
        lrow[c] = e;
        sum += e;
      }
      const float inv = 1.0f / sum;
      float* oprob = out + OUT_PROBS + (size_t)rr * C_;
      float bestp = -1.0f; int bestc = 1;
      for (int c = 0; c < C_; ++c) {
        float pr = lrow[c] * inv;
        oprob[c] = pr;
        if (c >= 1 && pr > bestp) { bestp = pr; bestc = c; }
      }
      const int bb = rr / P_, ppi = rr % P_;
      const int si = conn[(bb * 2 + 0) * P_ + ppi];
      const int oi = conn[(bb * 2 + 1) * P_ + ppi];
      const float osc = bestp * scores[bb * N_ + si] * scores[bb * N_ + oi];
      ov[rr] = osc;
      pp[rr] = bestp;
      pl[rr] = (float)bestc;
    }
  }
}

// ---------------- Kernel 3: per-batch iterative top-100 ---------------------
__global__ __launch_bounds__(256) void k_topk(
    float* __restrict__ ov, const float* __restrict__ pp,
    const float* __restrict__ pl, float* __restrict__ out) {
  __shared__ float sv[256];
  __shared__ int   si[256];
  const int b = blockIdx.x, tid = threadIdx.x;
  float* o = ov + b * P_;
  for (int t = 0; t < TOPK_; ++t) {
    float bv = -2.0f; int bi = 0;
    for (int p = tid; p < P_; p += 256) {
      float v = o[p];
      if (v > bv) { bv = v; bi = p; }
    }
    sv[tid] = bv; si[tid] = bi;
    __syncthreads();
    for (int s = 128; s > 0; s >>= 1) {
      if (tid < s) {
        float v2 = sv[tid + s]; int i2 = si[tid + s];
        if (v2 > sv[tid] || (v2 == sv[tid] && i2 < si[tid])) {
          sv[tid] = v2; si[tid] = i2;
        }
      }
      __syncthreads();
    }
    if (tid == 0) {
      const int bi0 = si[0];
      out[OUT_TVAL  + b * TOPK_ + t] = sv[0];
      out[OUT_TIDX  + b * TOPK_ + t] = (float)bi0;
      out[OUT_TLAB  + b * TOPK_ + t] = pl[b * P_ + bi0];
      out[OUT_TPROB + b * TOPK_ + t] = pp[b * P_ + bi0];
      o[bi0] = -1.0f;  // remove; ov fully rewritten by k_gemm2 each call
    }
    __threadfence();
    __syncthreads();
  }
}

extern "C" void kernel_launch(void* const* d_in, const int* in_sizes, int n_in,
                              void* d_out, int out_size, void* d_ws,
                              size_t ws_size, hipStream_t stream) {
  const float* inst   = (const float*)d_in[0];
  const float* phr    = (const float*)d_in[1];
  const float* scores = (const float*)d_in[2];
  const float* W1     = (const float*)d_in[3];
  const float* b1     = (const float*)d_in[4];
  const float* W2     = (const float*)d_in[5];
  const float* b2     = (const float*)d_in[6];
  const int*   conn   = (const int*)d_in[7];
  const int*   pidx   = (const int*)d_in[8];
  float* out = (float*)d_out;

  float* ws  = (float*)d_ws;
  float* h   = ws;                         // M_*H_ floats (324 MB)
  float* W2p = h + (size_t)M_ * H_;        // H_*64
  float* ov  = W2p + (size_t)H_ * CPAD_;   // M_
  float* pp  = ov + M_;                    // M_
  float* pl  = pp + M_;                    // M_

  k_pad_w2<<<(H_ * CPAD_ + 255) / 256, 256, 0, stream>>>(W2, W2p);

  dim3 g1((M_ + 127) / 128, 2);
  k_gemm1<<<g1, 512, 0, stream>>>(inst, phr, W1, b1, conn, pidx, h);

  k_gemm2<<<(M_ + 127) / 128, 256, 0, stream>>>(h, W2p, b2, scores, conn, out,
                                                ov, pp, pl);

  k_topk<<<B_, 256, 0, stream>>>(ov, pp, pl, out);
}